// Experts_74371653697640
// MI455X (gfx1250) — compile-verified
//
#include <hip/hip_runtime.h>

// Problem constants (match the reference).
#define NE 64    // experts
#define ND 512   // model dim
#define NH 1024  // hidden dim
#define NT 256   // tokens

typedef float v2f __attribute__((ext_vector_type(2)));
typedef float v4f __attribute__((ext_vector_type(4)));
typedef float v8f __attribute__((ext_vector_type(8)));

// ---------------- Pass 1: bucket tokens by expert ----------------

__global__ void zero_counts_kernel(int* counts) {
  if (threadIdx.x < NE) counts[threadIdx.x] = 0;
}

__global__ void bucket_kernel(const int* __restrict__ expert_idx,
                              int* __restrict__ counts,
                              int* __restrict__ lists) {
  int t = blockIdx.x * blockDim.x + threadIdx.x;
  if (t < NT) {
    int e = expert_idx[t];
    int slot = atomicAdd(&counts[e], 1);
    lists[e * NT + slot] = t;
  }
}

// ---------------- Pass 2: grouped expert MLP with f32 WMMA ----------------
//
// grid = (NE, NT/16). block = 256 threads = 8 waves (wave32).
// Each block: one expert, one tile of up to 16 tokens.
//
// WMMA operand layouts (wave32, V_WMMA_F32_16X16X4_F32):
//   A:   lane L holds A[L&15][khalf + v], khalf = (L<16 ? 0 : 2), v = 0..1
//   B:   lane L holds B[khalf + v][L&15]
//   C/D: lane L holds D[r + (L<16 ? 0 : 8)][L&15], r = 0..7
//
// Staged operands live in LDS as ROW PAIRS: P[k/2][n] = {M[k][n], M[k+1][n]}
// so one wmma's B operand is a single aligned ds_load_b64. Each wave runs TWO
// M-tiles per K-loop, sharing the B load between two wmmas (halves DS traffic,
// doubles in-flight weight loads). hP sits at LDS offset 0 so the long fc2
// K-loop can fold DS offsets into immediates.

__global__ __launch_bounds__(256) void moe_wmma_kernel(
    const float* __restrict__ x, const float* __restrict__ fc1_w,
    const float* __restrict__ fc2_w, const int* __restrict__ counts,
    const int* __restrict__ lists, float* __restrict__ out) {
  // Manual LDS layout: hP [0,64KB) | xP [64KB,96KB) | toks
  __shared__ __align__(16) v2f smem[(NH / 2 + ND / 2) * 16 + 8];
  v2f* const hP = smem;                      // pairs of h^T rows
  v2f* const xP = smem + (NH / 2) * 16;      // pairs of x^T rows
  int* const toks = (int*)(smem + (NH / 2 + ND / 2) * 16);

  const int e = blockIdx.x;
  const int g = blockIdx.y;
  const int c = counts[e];
  if (g * 16 >= c) return;  // uniform per block: EXEC stays all-1s below
  int nTok = c - g * 16;
  if (nTok > 16) nTok = 16;

  const int tid = threadIdx.x;
  const int lane = tid & 31;
  const int wave = tid >> 5;
  const int row = lane & 15;              // N (token slot) / A-row within half-wave
  const int kh2 = (lane < 16) ? 0 : 1;    // K-pair offset for this half-wave
  const int khalf = kh2 * 2;              // K offset (0 or 2)
  const int mrow = (lane < 16) ? 0 : 8;   // M offset for C/D rows

  if (tid < 16) toks[tid] = (tid < nTok) ? lists[e * NT + g * 16 + tid] : 0;
  __syncthreads();

  // Stage x^T tile into LDS as row pairs (pad missing tokens with zeros).
  // Each iteration: one global_load_b128 -> two ds_store_b64.
  for (int i = tid; i < (ND / 4) * 16; i += 256) {
    int n = i & 15;
    int q = i >> 4;  // quad index: covers x columns 4q..4q+3
    v4f v = {0.0f, 0.0f, 0.0f, 0.0f};
    if (n < nTok) v = *(const v4f*)(x + (size_t)toks[n] * ND + 4 * q);
    v2f lo = {v.x, v.y}, hi = {v.z, v.w};
    xP[(2 * q + 0) * 16 + n] = lo;
    xP[(2 * q + 1) * 16 + n] = hi;
  }
  __syncthreads();

  // ---- fc1: h = silu(x @ W1^T), W1 is [H, D] row-major ----
  const float* w1e = fc1_w + (size_t)e * NH * ND;
  for (int mt = wave * 2; mt < NH / 16; mt += 16) {  // tiles mt and mt+1
    const int jBase = mt * 16;
    v8f acc0 = {}, acc1 = {};
    const float* arow0 = w1e + (size_t)(jBase + row) * ND + khalf;
    const float* arow1 = arow0 + (size_t)16 * ND;
    const v2f* bcol = xP + kh2 * 16 + row;
#pragma unroll 8
    for (int kb = 0; kb < ND / 4; ++kb) {
      v2f b = bcol[kb * 32];  // pair (2*kb + kh2), column `row` (shared)
      v2f a0 = *(const v2f*)(arow0 + kb * 4);
      v2f a1 = *(const v2f*)(arow1 + kb * 4);
      acc0 = __builtin_amdgcn_wmma_f32_16x16x4_f32(false, a0, false, b,
                                                   (short)0, acc0, false, false);
      acc1 = __builtin_amdgcn_wmma_f32_16x16x4_f32(false, a1, false, b,
                                                   (short)0, acc1, false, false);
    }
    // SiLU (fast hardware rcp) and store as row pairs: ds_store_b64 x8.
#pragma unroll
    for (int r = 0; r < 8; r += 2) {
      v2f s0, s1;
      s0.x = acc0[r] * __builtin_amdgcn_rcpf(1.0f + __expf(-acc0[r]));
      s0.y = acc0[r + 1] * __builtin_amdgcn_rcpf(1.0f + __expf(-acc0[r + 1]));
      s1.x = acc1[r] * __builtin_amdgcn_rcpf(1.0f + __expf(-acc1[r]));
      s1.y = acc1[r + 1] * __builtin_amdgcn_rcpf(1.0f + __expf(-acc1[r + 1]));
      hP[((jBase + mrow + r) >> 1) * 16 + row] = s0;
      hP[((jBase + 16 + mrow + r) >> 1) * 16 + row] = s1;
    }
  }
  __syncthreads();

  // ---- fc2: out = h @ W2^T, W2 is [D, H] row-major ----
  const float* w2e = fc2_w + (size_t)e * ND * NH;
  for (int mt = wave * 2; mt < ND / 16; mt += 16) {  // tiles mt and mt+1
    const int dBase = mt * 16;
    v8f acc0 = {}, acc1 = {};
    const float* arow0 = w2e + (size_t)(dBase + row) * NH + khalf;
    const float* arow1 = arow0 + (size_t)16 * NH;
    const v2f* bcol = hP + kh2 * 16 + row;
#pragma unroll 8
    for (int kb = 0; kb < NH / 4; ++kb) {
      v2f b = bcol[kb * 32];
      v2f a0 = *(const v2f*)(arow0 + kb * 4);
      v2f a1 = *(const v2f*)(arow1 + kb * 4);
      acc0 = __builtin_amdgcn_wmma_f32_16x16x4_f32(false, a0, false, b,
                                                   (short)0, acc0, false, false);
      acc1 = __builtin_amdgcn_wmma_f32_16x16x4_f32(false, a1, false, b,
                                                   (short)0, acc1, false, false);
    }
    if (row < nTok) {
      float* o = out + (size_t)toks[row] * ND + dBase + mrow;
      v4f o0 = {acc0[0], acc0[1], acc0[2], acc0[3]};
      v4f o1 = {acc0[4], acc0[5], acc0[6], acc0[7]};
      v4f o2 = {acc1[0], acc1[1], acc1[2], acc1[3]};
      v4f o3 = {acc1[4], acc1[5], acc1[6], acc1[7]};
      *(v4f*)(o) = o0;         // global_store_b128
      *(v4f*)(o + 4) = o1;
      *(v4f*)(o + 16) = o2;    // tile mt+1
      *(v4f*)(o + 20) = o3;
    }
  }
}

// ---------------- Launch ----------------

extern "C" void kernel_launch(void* const* d_in, const int* in_sizes, int n_in,
                              void* d_out, int out_size, void* d_ws, size_t ws_size,
                              hipStream_t stream) {
  const float* x = (const float*)d_in[0];
  const int* expert_idx = (const int*)d_in[1];
  const float* fc1_w = (const float*)d_in[2];
  const float* fc2_w = (const float*)d_in[3];
  float* out = (float*)d_out;

  // Workspace layout: counts[NE] | lists[NE*NT]   (~66 KB)
  int* counts = (int*)d_ws;
  int* lists = counts + NE;

  zero_counts_kernel<<<1, 64, 0, stream>>>(counts);
  bucket_kernel<<<1, 256, 0, stream>>>(expert_idx, counts, lists);

  dim3 grid(NE, NT / 16);
  moe_wmma_kernel<<<grid, 256, 0, stream>>>(x, fc1_w, fc2_w, counts, lists, out);
}